// ContinuousNormalizingFlow_83949430767894
// MI455X (gfx1250) — compile-verified
//
#include <hip/hip_runtime.h>

typedef float v2f __attribute__((ext_vector_type(2)));
typedef float v8f __attribute__((ext_vector_type(8)));

#define NSAMP   32768
#define HID     64
#define THREADS 128
#define WAVES   4
#define SPB     64            // samples per block (4 waves x 16)
#define NBLK    (NSAMP/SPB)   // 512 blocks
#define WST     72            // weight-slab row stride (8 mod 64 -> conflict-free B loads)
#define SST     68            // staging row stride   (4 mod 64 -> conflict-free A loads/C stores)

// D(16x16,f32) += A(16x4,f32) * B(4x16,f32)
__device__ __forceinline__ v8f wmma4(v2f a, v2f b, v8f c) {
  return __builtin_amdgcn_wmma_f32_16x16x4_f32(false, a, false, b, (short)0, c, false, false);
}

// Compiler-only reordering barrier for wave-private LDS staging.
// DS ops from one wave execute in order (ISA ch.11), so a RAW through LDS
// within the same wave needs no counter drain -- only the compiler must not
// reorder the loads before the stores. VGPR hazards on load results are
// handled by the compiler's own fine-grained s_wait_dscnt.
__device__ __forceinline__ void ldsfence() {
  asm volatile("" ::: "memory");
}

// sigmoid via v_exp_f32 + v_rcp_f32 (avoids the IEEE div_scale/div_fixup chain)
__device__ __forceinline__ float sigm(float x) {
#if __has_builtin(__builtin_amdgcn_rcpf)
  return __builtin_amdgcn_rcpf(1.0f + __expf(-x));
#else
  return 1.0f / (1.0f + __expf(-x));
#endif
}

__global__ __launch_bounds__(THREADS)
void cnf_wmma_kernel(const float* __restrict__ yin,
                     const float* __restrict__ eW0, const float* __restrict__ eb0,
                     const float* __restrict__ eW1, const float* __restrict__ eb1,
                     const float* __restrict__ eW2,
                     const float* __restrict__ dW0, const float* __restrict__ db0,
                     const float* __restrict__ dW1, const float* __restrict__ db1,
                     const float* __restrict__ dW2,
                     const int*   __restrict__ t1p,
                     float* __restrict__ out)
{
  // Weight slabs for the CURRENT phase (reloaded between phases).
  __shared__ __attribute__((aligned(16))) float sW1T[HID*WST]; // B fwd: [i][j] = W1[j][i]
  __shared__ __attribute__((aligned(16))) float sW1 [HID*WST]; // B bwd: [j][i] = W1[j][i]
  __shared__ __attribute__((aligned(16))) float sW0B[4*WST];   // z0 B: k<3 -> W0[n][k], k=3 -> b0[n]
  __shared__ __attribute__((aligned(16))) float sBf [HID*16];  // grad B: col0=W0[:,0], col1=W0[:,1]
  __shared__ float sB1[HID];
  __shared__ float sW2[HID];
  __shared__ float sW0r[HID*3];
  __shared__ __attribute__((aligned(16))) float sStg[WAVES][3*16*SST];
  __shared__ float sY[SPB][2];
  __shared__ float sLP[SPB];

  const int tid   = threadIdx.x;
  const int gbase = blockIdx.x * SPB;

  const float* W0s[2] = {eW0, dW0};
  const float* b0s[2] = {eb0, db0};
  const float* W1s[2] = {eW1, dW1};
  const float* b1s[2] = {eb1, db1};
  const float* W2s[2] = {eW2, dW2};
  const int nsteps[2] = {10, *t1p};
  const float dt = 1.0f;

  for (int idx = tid; idx < SPB; idx += THREADS) {
    sY[idx][0] = yin[(gbase + idx)*2 + 0];
    sY[idx][1] = yin[(gbase + idx)*2 + 1];
    sLP[idx]   = 0.0f;
  }

  const int lane = tid & 31;
  const int wave = tid >> 5;
  const int half = lane >> 4;     // K/M lane-half per WMMA layouts
  const int ln   = lane & 15;
  const int wm0  = wave * 16;
  float* stg0 = &sStg[wave][0];
  float* stg1 = stg0 + 16*SST;
  float* stg2 = stg1 + 16*SST;

  for (int p = 0; p < 2; ++p) {
    // ---- load this phase's weights into LDS ----
    __syncthreads();
    for (int idx = tid; idx < HID*HID; idx += THREADS) {
      int j = idx >> 6, i = idx & 63;
      float w = W1s[p][idx];              // W1[j][i]
      sW1 [j*WST + i] = w;
      sW1T[i*WST + j] = w;
    }
    for (int idx = tid; idx < 4*HID; idx += THREADS) {
      int k = idx >> 6, n = idx & 63;
      sW0B[k*WST + n] = (k < 3) ? W0s[p][n*3 + k] : b0s[p][n];
    }
    for (int idx = tid; idx < HID*16; idx += THREADS) {
      int j = idx >> 4, n = idx & 15;
      sBf[idx] = (n < 2) ? W0s[p][j*3 + n] : 0.0f;
    }
    for (int idx = tid; idx < HID; idx += THREADS) {
      sB1[idx] = b1s[p][idx];
      sW2[idx] = W2s[p][idx];
    }
    for (int idx = tid; idx < HID*3; idx += THREADS)
      sW0r[idx] = W0s[p][idx];
    __syncthreads();

    for (int step = 0; step < nsteps[p]; ++step) {
      const float tval = (float)step;   // t0=0, dt=1
      const v8f zc = {};

      // ---- A_h = [y0, y1, t, 1]  (16x4 A-fragment) ----
      float yy0 = sY[wm0 + ln][0];
      float yy1 = sY[wm0 + ln][1];
      v2f ah;
      ah.x = half ? tval : yy0;   // K = 0 / 2
      ah.y = half ? 1.0f : yy1;   // K = 1 / 3

      // ---- z0 = h @ W0^T + b0 : 4 WMMAs ----
      v8f z0[4], d0[4], e0[4];
      #pragma unroll
      for (int nc = 0; nc < 4; ++nc) {
        int k0 = 2*half, n = nc*16 + ln;
        v2f b; b.x = sW0B[k0*WST + n]; b.y = sW0B[(k0+1)*WST + n];
        z0[nc] = wmma4(ah, b, zc);
      }

      // ---- SiLU family on z0; stage a0, da0_0, da0_1 as A-inputs ----
      ldsfence();
      #pragma unroll
      for (int nc = 0; nc < 4; ++nc) {
        int j = nc*16 + ln;
        float w00 = sW0r[j*3 + 0];
        float w01 = sW0r[j*3 + 1];
        #pragma unroll
        for (int v = 0; v < 8; ++v) {
          float x  = z0[nc][v];
          float s  = sigm(x);
          float sp = s*(1.0f - s);
          float dd = s + x*sp;                        // silu'
          float ee = sp*(2.0f + x*(1.0f - 2.0f*s));   // silu''
          d0[nc][v] = dd;
          e0[nc][v] = ee;
          int row = (v + 8*half)*SST + j;
          stg0[row] = x*s;                            // a0
          stg1[row] = dd*w00;                         // da0 (tangent e0)
          stg2[row] = dd*w01;                         // da0 (tangent e1)
        }
      }
      ldsfence();

      // ---- forward family: z1 / dz1_0 / dz1_1 = {.} @ W1^T  (192 WMMAs) ----
      v8f z1[4], dz10[4], dz11[4];
      {
        // kc = 0 peeled: C = inline 0 (no accumulator zero-init movs)
        int arow = ln*SST + 2*half;
        v2f A0 = *(const v2f*)&stg0[arow];
        v2f A1 = *(const v2f*)&stg1[arow];
        v2f A2 = *(const v2f*)&stg2[arow];
        int k0 = 2*half;
        #pragma unroll
        for (int nc = 0; nc < 4; ++nc) {
          int n = nc*16 + ln;
          v2f B; B.x = sW1T[k0*WST + n]; B.y = sW1T[(k0+1)*WST + n];
          z1[nc]   = wmma4(A0, B, zc);
          dz10[nc] = wmma4(A1, B, zc);
          dz11[nc] = wmma4(A2, B, zc);
        }
      }
      for (int kc = 1; kc < 16; ++kc) {
        int arow = ln*SST + kc*4 + 2*half;
        v2f A0 = *(const v2f*)&stg0[arow];
        v2f A1 = *(const v2f*)&stg1[arow];
        v2f A2 = *(const v2f*)&stg2[arow];
        int k0 = kc*4 + 2*half;
        #pragma unroll
        for (int nc = 0; nc < 4; ++nc) {
          int n = nc*16 + ln;
          v2f B; B.x = sW1T[k0*WST + n]; B.y = sW1T[(k0+1)*WST + n];
          z1[nc]   = wmma4(A0, B, z1[nc]);
          dz10[nc] = wmma4(A1, B, dz10[nc]);
          dz11[nc] = wmma4(A2, B, dz11[nc]);
        }
      }

      // ---- SiLU family on z1; stage v1 = w2*silu'(z1) and its 2 tangents ----
      ldsfence();
      #pragma unroll
      for (int nc = 0; nc < 4; ++nc) {
        int j = nc*16 + ln;
        float b1v = sB1[j];
        float w2v = sW2[j];
        #pragma unroll
        for (int v = 0; v < 8; ++v) {
          float x  = z1[nc][v] + b1v;
          float s  = sigm(x);
          float sp = s*(1.0f - s);
          float dd = s + x*sp;
          float ee = sp*(2.0f + x*(1.0f - 2.0f*s));
          int row = (v + 8*half)*SST + j;
          stg0[row] = w2v*dd;                 // dphi/dz1
          stg1[row] = w2v*ee*dz10[nc][v];     // its tangent (e0)
          stg2[row] = w2v*ee*dz11[nc][v];     // its tangent (e1)
        }
      }
      ldsfence();

      // ---- backward family: g0 / dg0_0 / dg0_1 = {.} @ W1  (192 WMMAs) ----
      v8f g0[4], dg00[4], dg01[4];
      {
        int arow = ln*SST + 2*half;
        v2f A0 = *(const v2f*)&stg0[arow];
        v2f A1 = *(const v2f*)&stg1[arow];
        v2f A2 = *(const v2f*)&stg2[arow];
        int k0 = 2*half;
        #pragma unroll
        for (int nc = 0; nc < 4; ++nc) {
          int n = nc*16 + ln;
          v2f B; B.x = sW1[k0*WST + n]; B.y = sW1[(k0+1)*WST + n];
          g0[nc]   = wmma4(A0, B, zc);
          dg00[nc] = wmma4(A1, B, zc);
          dg01[nc] = wmma4(A2, B, zc);
        }
      }
      for (int kc = 1; kc < 16; ++kc) {
        int arow = ln*SST + kc*4 + 2*half;
        v2f A0 = *(const v2f*)&stg0[arow];
        v2f A1 = *(const v2f*)&stg1[arow];
        v2f A2 = *(const v2f*)&stg2[arow];
        int k0 = kc*4 + 2*half;
        #pragma unroll
        for (int nc = 0; nc < 4; ++nc) {
          int n = nc*16 + ln;
          v2f B; B.x = sW1[k0*WST + n]; B.y = sW1[(k0+1)*WST + n];
          g0[nc]   = wmma4(A0, B, g0[nc]);
          dg00[nc] = wmma4(A1, B, dg00[nc]);
          dg01[nc] = wmma4(A2, B, dg01[nc]);
        }
      }

      // ---- q = g0*silu'(z0); u_d = dg0_d*silu'(z0) + g0*silu''(z0)*W0[:,d] ----
      ldsfence();
      #pragma unroll
      for (int nc = 0; nc < 4; ++nc) {
        int j = nc*16 + ln;
        float w00 = sW0r[j*3 + 0];
        float w01 = sW0r[j*3 + 1];
        #pragma unroll
        for (int v = 0; v < 8; ++v) {
          float dd = d0[nc][v], ee = e0[nc][v], g = g0[nc][v];
          int row = (v + 8*half)*SST + j;
          stg0[row] = g*dd;                           // q = dphi/dz0
          stg1[row] = dg00[nc][v]*dd + g*ee*w00;      // u0
          stg2[row] = dg01[nc][v]*dd + g*ee*w01;      // u1
        }
      }
      ldsfence();

      // ---- contractions with W0 columns: f and Laplacian (48 WMMAs) ----
      v8f Df, Du0, Du1;
      {
        int arow = ln*SST + 2*half;
        v2f A0 = *(const v2f*)&stg0[arow];
        v2f A1 = *(const v2f*)&stg1[arow];
        v2f A2 = *(const v2f*)&stg2[arow];
        int k0 = 2*half;
        v2f B; B.x = sBf[k0*16 + ln]; B.y = sBf[(k0+1)*16 + ln];
        Df  = wmma4(A0, B, zc);
        Du0 = wmma4(A1, B, zc);
        Du1 = wmma4(A2, B, zc);
      }
      for (int kc = 1; kc < 16; ++kc) {
        int arow = ln*SST + kc*4 + 2*half;
        v2f A0 = *(const v2f*)&stg0[arow];
        v2f A1 = *(const v2f*)&stg1[arow];
        v2f A2 = *(const v2f*)&stg2[arow];
        int k0 = kc*4 + 2*half;
        v2f B; B.x = sBf[k0*16 + ln]; B.y = sBf[(k0+1)*16 + ln];
        Df  = wmma4(A0, B, Df);    // col0 = f0, col1 = f1
        Du0 = wmma4(A1, B, Du0);   // col0 = d f0/d y0
        Du1 = wmma4(A2, B, Du1);   // col1 = d f1/d y1
      }

      // ---- Euler update of (y, logp) held in LDS ----
      ldsfence();
      #pragma unroll
      for (int v = 0; v < 8; ++v) {
        float trp = (ln == 0) ? Du0[v] : Du1[v];
        float tr  = trp + __shfl_xor(trp, 1, 32);  // lane0: H00+H11
        int m = wm0 + v + 8*half;
        if (ln < 2)  sY[m][ln] += dt * Df[v];
        if (ln == 0) sLP[m]    += dt * tr;
      }
      ldsfence();
    }
  }

  // ---- write out (y, logp) ----
  __syncthreads();
  for (int idx = tid; idx < SPB; idx += THREADS) {
    out[(gbase + idx)*2 + 0] = sY[idx][0];
    out[(gbase + idx)*2 + 1] = sY[idx][1];
    out[NSAMP*2 + gbase + idx] = sLP[idx];
  }
}

extern "C" void kernel_launch(void* const* d_in, const int* in_sizes, int n_in,
                              void* d_out, int out_size, void* d_ws, size_t ws_size,
                              hipStream_t stream) {
  (void)in_sizes; (void)n_in; (void)out_size; (void)d_ws; (void)ws_size;
  const float* y   = (const float*)d_in[0];
  const float* eW0 = (const float*)d_in[1];
  const float* eb0 = (const float*)d_in[2];
  const float* eW1 = (const float*)d_in[3];
  const float* eb1 = (const float*)d_in[4];
  const float* eW2 = (const float*)d_in[5];
  const float* dW0 = (const float*)d_in[7];
  const float* db0 = (const float*)d_in[8];
  const float* dW1 = (const float*)d_in[9];
  const float* db1 = (const float*)d_in[10];
  const float* dW2 = (const float*)d_in[11];
  const int*   t1  = (const int*)d_in[13];
  float* out = (float*)d_out;
  cnf_wmma_kernel<<<NBLK, THREADS, 0, stream>>>(y, eW0, eb0, eW1, eb1, eW2,
                                                dW0, db0, dW1, db1, dW2, t1, out);
}